// CustomGCN_57492432224710
// MI455X (gfx1250) — compile-verified
//
#include <hip/hip_runtime.h>
#include <math.h>

typedef float v2f __attribute__((ext_vector_type(2)));
typedef float v8f __attribute__((ext_vector_type(8)));

#define HDIM 128
#define ROWS_PER_BLOCK 64
#define XSTRIDE 132   // 128 + 4 pad: lanes 0..15 hit distinct LDS banks

// ---------------- degree / norm kernels ----------------
__global__ __launch_bounds__(256) void deg_init_kernel(float* deg, int n) {
  int i = blockIdx.x * 256 + threadIdx.x;
  if (i < n) deg[i] = 1.0f;                       // self-loop contribution
}

__global__ __launch_bounds__(256) void deg_count_kernel(const int* __restrict__ dst,
                                                        float* deg, int e) {
  int i = blockIdx.x * 256 + threadIdx.x;
  if (i < e) atomicAdd(&deg[dst[i]], 1.0f);
}

__global__ __launch_bounds__(256) void deg_rsqrt_kernel(float* deg, int n) {
  int i = blockIdx.x * 256 + threadIdx.x;
  if (i < n) deg[i] = rsqrtf(deg[i]);             // deg >= 1 always (self-loop)
}

// ---------------- fp32 WMMA GEMM: Y[nrows,128] = (relu?)X[nrows,128] @ W[128,128] ----------------
__global__ __launch_bounds__(256) void gemm128_wmma_kernel(
    const float* __restrict__ X, const float* __restrict__ W,
    float* __restrict__ Y, int nrows, int relu_input)
{
  __shared__ float lds_x[ROWS_PER_BLOCK * XSTRIDE];  // 33792 B
  __shared__ float lds_w[HDIM * HDIM];               // 65536 B
  const int t = threadIdx.x;
  const int row0 = blockIdx.x * ROWS_PER_BLOCK;

  // Stage W (128x128) into LDS, float4 per transfer
  for (int i = t; i < HDIM * HDIM / 4; i += 256) {
    *(float4*)&lds_w[i * 4] = ((const float4*)W)[i];
  }
  // Stage X tile (64 rows x 128) with padded stride; fuse optional ReLU
  for (int i = t; i < ROWS_PER_BLOCK * (HDIM / 4); i += 256) {
    int r  = i >> 5;     // 32 float4 per row
    int c4 = i & 31;
    int gr = row0 + r;
    float4 v = make_float4(0.f, 0.f, 0.f, 0.f);
    if (gr < nrows) v = ((const float4*)(X + (size_t)gr * HDIM))[c4];
    if (relu_input) {
      v.x = fmaxf(v.x, 0.f); v.y = fmaxf(v.y, 0.f);
      v.z = fmaxf(v.z, 0.f); v.w = fmaxf(v.w, 0.f);
    }
    *(float4*)&lds_x[r * XSTRIDE + c4 * 4] = v;
  }
  __syncthreads();

  const int wave  = t >> 5;             // 8 waves -> 8 col stripes of 16
  const int lane  = t & 31;
  const int lm    = lane & 15;
  const int khalf = (lane >> 4) << 1;   // lanes 16..31 hold K+2,K+3 halves
  const int col0  = wave * 16;

  for (int rt = 0; rt < ROWS_PER_BLOCK / 16; ++rt) {
    v8f acc = {0.f, 0.f, 0.f, 0.f, 0.f, 0.f, 0.f, 0.f};
    const float* xrow = &lds_x[(rt * 16 + lm) * XSTRIDE];
    #pragma unroll
    for (int k = 0; k < HDIM; k += 4) {
      // A fragment: 16x4 f32, row M=lm, K = k+khalf, k+khalf+1 (8B-aligned ds_load_b64)
      v2f a = *(const v2f*)&xrow[k + khalf];
      // B fragment: 4x16 f32, rows K=k+khalf, k+khalf+1, col = col0+lm
      v2f b;
      b.x = lds_w[(k + khalf)     * HDIM + col0 + lm];
      b.y = lds_w[(k + khalf + 1) * HDIM + col0 + lm];
      acc = __builtin_amdgcn_wmma_f32_16x16x4_f32(false, a, false, b,
                                                  (short)0, acc, false, false);
    }
    // C/D layout: VGPR j -> (M=j, N=lane) lanes 0..15 ; (M=j+8, N=lane-16) lanes 16..31
    const int gcol  = col0 + lm;
    const int rbase = row0 + rt * 16 + ((lane >> 4) << 3);
    #pragma unroll
    for (int j = 0; j < 8; ++j) {
      int gr = rbase + j;
      if (gr < nrows) Y[(size_t)gr * HDIM + gcol] = acc[j];
    }
  }
}

// ---------------- aggregation: out[n,:] = bias + xw[n,:]*dinv[n]^2 (self loop) ----------------
__global__ __launch_bounds__(256) void agg_init_kernel(
    const float* __restrict__ xw, const float* __restrict__ dinv,
    const float* __restrict__ bias, float* __restrict__ out, int total)
{
  int i = blockIdx.x * 256 + threadIdx.x;
  if (i >= total) return;
  int node = i >> 7;        // H = 128
  int h    = i & 127;
  float d = dinv[node];
  out[i] = bias[h] + xw[i] * d * d;
}

// ---------------- edge scatter: out[dst,:] += xw[src,:] * dinv[src]*dinv[dst] ----------------
__global__ __launch_bounds__(256) void agg_scatter_kernel(
    const float* __restrict__ xw, const float* __restrict__ dinv,
    const int* __restrict__ src, const int* __restrict__ dst,
    float* __restrict__ out, int e)
{
  int wid  = (blockIdx.x * 256 + threadIdx.x) >> 5;   // one wave32 per edge
  int lane = threadIdx.x & 31;
  if (wid >= e) return;
  int s = src[wid], d = dst[wid];
  float norm = dinv[s] * dinv[d];
  const float* xs = xw + (size_t)s * HDIM;
  float* od = out + (size_t)d * HDIM;
  #pragma unroll
  for (int j = 0; j < 4; ++j) {
    int h = lane + j * 32;                            // coalesced 128B groups
    atomicAdd(&od[h], xs[h] * norm);
  }
}

// ---------------- readout coefficients: c0 = be.Wl2 + bl ; c1 = We.Wl2 ----------------
__global__ __launch_bounds__(128) void coef_kernel(
    const float* __restrict__ We, const float* __restrict__ be,
    const float* __restrict__ Wl, const float* __restrict__ bl, float* c)
{
  __shared__ float s0[HDIM];
  __shared__ float s1[HDIM];
  int t = threadIdx.x;
  float wl2 = Wl[HDIM + t];
  s0[t] = be[t] * wl2;
  s1[t] = We[t] * wl2;
  __syncthreads();
  for (int s = 64; s > 0; s >>= 1) {
    if (t < s) { s0[t] += s0[t + s]; s1[t] += s1[t + s]; }
    __syncthreads();
  }
  if (t == 0) { c[0] = s0[0] + bl[0]; c[1] = s1[0]; }
}

// ---------------- readout: out[e] = h2[src[e]].Wl[:128] + ea[e]*c1 + c0 ----------------
__global__ __launch_bounds__(256) void readout_kernel(
    const float* __restrict__ h2, const int* __restrict__ src,
    const float* __restrict__ ea, const float* __restrict__ Wl,
    const float* __restrict__ c, float* __restrict__ out, int e)
{
  int wid  = (blockIdx.x * 256 + threadIdx.x) >> 5;   // one wave32 per edge
  int lane = threadIdx.x & 31;
  if (wid >= e) return;
  int s = src[wid];
  float4 hv = ((const float4*)(h2 + (size_t)s * HDIM))[lane];
  float4 wv = ((const float4*)Wl)[lane];
  float sum = hv.x * wv.x + hv.y * wv.y + hv.z * wv.z + hv.w * wv.w;
  #pragma unroll
  for (int off = 16; off > 0; off >>= 1) sum += __shfl_xor(sum, off, 32);
  if (lane == 0) out[wid] = sum + ea[wid] * c[1] + c[0];
}

// ---------------- host launch ----------------
extern "C" void kernel_launch(void* const* d_in, const int* in_sizes, int n_in,
                              void* d_out, int out_size, void* d_ws, size_t ws_size,
                              hipStream_t stream) {
  const float* x   = (const float*)d_in[0];
  const int*   ei  = (const int*)  d_in[1];
  const float* ea  = (const float*)d_in[2];
  const float* W1  = (const float*)d_in[3];
  const float* b1  = (const float*)d_in[4];
  const float* W2  = (const float*)d_in[5];
  const float* b2  = (const float*)d_in[6];
  const float* We  = (const float*)d_in[7];
  const float* be  = (const float*)d_in[8];
  const float* Wl  = (const float*)d_in[9];
  const float* bl  = (const float*)d_in[10];
  float* out = (float*)d_out;

  const int N = in_sizes[0] / HDIM;
  const int E = in_sizes[2];
  const int* src = ei;
  const int* dst = ei + E;

  // workspace layout (floats): dinv[N] | c[8 pad] | bufA[N*128] | bufB[N*128]
  float* ws    = (float*)d_ws;
  float* dinv  = ws;
  float* coefs = ws + N;
  float* bufA  = ws + N + 8;
  float* bufB  = bufA + (size_t)N * HDIM;

  const int nB    = (N + 255) / 256;
  const int eB    = (E + 255) / 256;
  const int nhB   = ((size_t)N * HDIM + 255) / 256;
  const int edgeB = (E + 7) / 8;               // 8 waves per block, 1 edge per wave
  const int gemmB = (N + ROWS_PER_BLOCK - 1) / ROWS_PER_BLOCK;

  // degrees -> dinv
  deg_init_kernel <<<nB, 256, 0, stream>>>(dinv, N);
  deg_count_kernel<<<eB, 256, 0, stream>>>(dst, dinv, E);
  deg_rsqrt_kernel<<<nB, 256, 0, stream>>>(dinv, N);

  // layer 1: xw1 = x @ W1 ; out1 = b1 + self-loop + scatter
  gemm128_wmma_kernel<<<gemmB, 256, 0, stream>>>(x, W1, bufA, N, 0);
  agg_init_kernel   <<<nhB,   256, 0, stream>>>(bufA, dinv, b1, bufB, N * HDIM);
  agg_scatter_kernel<<<edgeB, 256, 0, stream>>>(bufA, dinv, src, dst, bufB, E);

  // layer 2: xw2 = relu(out1) @ W2 (ReLU fused into GEMM load) ; out2 aggregation
  gemm128_wmma_kernel<<<gemmB, 256, 0, stream>>>(bufB, W2, bufA, N, 1);
  agg_init_kernel   <<<nhB,   256, 0, stream>>>(bufA, dinv, b2, bufB, N * HDIM);
  agg_scatter_kernel<<<edgeB, 256, 0, stream>>>(bufA, dinv, src, dst, bufB, E);

  // edge readout with folded edge-feature branch
  coef_kernel   <<<1,     128, 0, stream>>>(We, be, Wl, bl, coefs);
  readout_kernel<<<edgeB, 256, 0, stream>>>(bufB, src, ea, Wl, coefs, out, E);
}